// DynamicAccessor_86260123174136
// MI455X (gfx1250) — compile-verified
//
#include <hip/hip_runtime.h>
#include <hip/hip_bf16.h>
#include <math.h>

#define Bsz 32
#define Hd  1024
#define Nn  1024
#define Wd  128
#define NRh 4
#define EPSv 1e-6f

typedef float v2f __attribute__((ext_vector_type(2)));
typedef float v8f __attribute__((ext_vector_type(8)));

// ---- workspace layout (float offsets) ----
#define OFF_WKEY   0u          // B*128
#define OFF_ERASE  4096u       // B*128
#define OFF_ADD    8192u       // B*128
#define OFF_RKEY   12288u      // B*512
#define OFF_SMALL  28672u      // 1024: wbeta[32] | walloc@32 | wwrite@64 | rbeta@128(128) | freeg@256(128) | piRaw@384(384)
#define OFF_U      29696u      // B*N
#define OFF_WC     62464u      // B*N
#define OFF_WLW    95232u      // B*N
#define OFF_RC     128000u     // B*4*N
#define OFF_FWD    259072u     // B*4*N
#define OFF_BWD    390144u     // B*4*N
#define OFF_WLREAD 521216u     // B*4*N
#define OFF_MEMNEW 652288u     // B*N*W

__device__ __forceinline__ float sigm_(float x){ return 1.0f/(1.0f+expf(-x)); }
__device__ __forceinline__ float softplus_(float x){ return x > 20.0f ? x : log1pf(expf(x)); }

// ---------------- zero bwd accumulator ----------------
__global__ void k_zero(float* ws){
  int id = blockIdx.x*256 + threadIdx.x;
  if (id < Bsz*NRh*Nn) ws[OFF_BWD + id] = 0.0f;
}

// ---------------- big projections via f32 WMMA 16x16x4 ----------------
__global__ __launch_bounds__(32) void k_proj_wmma(const float* __restrict__ hidden,
    const float* __restrict__ Wkey, const float* __restrict__ Werase,
    const float* __restrict__ Wadd, const float* __restrict__ Wrkey,
    float* __restrict__ ws){
  int bid = blockIdx.x;
  const float* Wm; float* out; int od, act, local;
  if (bid < 16)      { Wm=Wkey;   out=ws+OFF_WKEY;  od=128; act=0; local=bid; }
  else if (bid < 32) { Wm=Werase; out=ws+OFF_ERASE; od=128; act=1; local=bid-16; }
  else if (bid < 48) { Wm=Wadd;   out=ws+OFF_ADD;   od=128; act=0; local=bid-32; }
  else               { Wm=Wrkey;  out=ws+OFF_RKEY;  od=512; act=0; local=bid-48; }
  int ctiles = od >> 4;
  int rt = local / ctiles, ct = local % ctiles;
  int row0 = rt*16, col0 = ct*16;
  int lane = threadIdx.x, lo = lane & 15, hi = lane >> 4;
  const float* Arow = hidden + (size_t)(row0+lo)*Hd;
  const float* Bp   = Wm + col0 + lo;
  v8f acc = {0,0,0,0,0,0,0,0};
  for (int k = 0; k < Hd; k += 4) {
    int ka = k + 2*hi;
    v2f a, b;
    a.x = Arow[ka];               a.y = Arow[ka+1];
    b.x = Bp[(size_t)ka*od];      b.y = Bp[(size_t)(ka+1)*od];
    acc = __builtin_amdgcn_wmma_f32_16x16x4_f32(false, a, false, b, (short)0, acc, false, false);
  }
  for (int v = 0; v < 8; ++v) {
    int m = v + 8*hi;
    float val = acc[v];
    val = (act == 0) ? tanhf(val) : sigm_(val);
    out[(size_t)(row0+m)*od + col0 + lo] = val;
  }
}

// ---------------- small projections (betas/gates/pi logits) ----------------
__global__ void k_proj_small(const float* __restrict__ hidden,
    const float* __restrict__ Wbeta, const float* __restrict__ Walloc,
    const float* __restrict__ Wwrite, const float* __restrict__ Wrbeta,
    const float* __restrict__ Wrfree, const float* __restrict__ Wrpi,
    float* __restrict__ ws){
  __shared__ float red[256];
  int job = blockIdx.x, b = job / 23, j = job % 23;
  const float* Wm; int od, col, act; float* smalls = ws + OFF_SMALL; float* out;
  if (j == 0)      { Wm=Wbeta;  od=1;  col=0;    act=2; out=smalls + b; }
  else if (j == 1) { Wm=Walloc; od=1;  col=0;    act=1; out=smalls + 32 + b; }
  else if (j == 2) { Wm=Wwrite; od=1;  col=0;    act=1; out=smalls + 64 + b; }
  else if (j < 7)  { Wm=Wrbeta; od=4;  col=j-3;  act=2; out=smalls + 128 + b*4 + (j-3); }
  else if (j < 11) { Wm=Wrfree; od=4;  col=j-7;  act=1; out=smalls + 256 + b*4 + (j-7); }
  else             { Wm=Wrpi;   od=12; col=j-11; act=3; out=smalls + 384 + b*12 + (j-11); }
  float s = 0.0f;
  const float* hb = hidden + (size_t)b*Hd;
  for (int k = threadIdx.x; k < Hd; k += 256) s += hb[k] * Wm[(size_t)k*od + col];
  red[threadIdx.x] = s; __syncthreads();
  for (int o = 128; o > 0; o >>= 1) { if (threadIdx.x < o) red[threadIdx.x] += red[threadIdx.x+o]; __syncthreads(); }
  if (threadIdx.x == 0) {
    float v = red[0];
    if (act == 1) v = sigm_(v);
    else if (act == 2) v = softplus_(v);
    *out = v;
  }
}

// ---------------- usage update ----------------
__global__ void k_usage(const float* __restrict__ usage, const float* __restrict__ wlpw,
                        const float* __restrict__ wlpr, float* __restrict__ ws){
  int id = blockIdx.x*256 + threadIdx.x;  // B*N
  int b = id >> 10, n = id & 1023;
  const float* freeg = ws + OFF_SMALL + 256;
  float u0 = usage[id];
  float u = u0 + (1.0f - u0) * wlpw[id];  // NW==1
  float psi = 1.0f;
  for (int r = 0; r < NRh; ++r)
    psi *= (1.0f - freeg[b*4+r] * wlpr[((size_t)b*4+r)*Nn + n]);
  ws[OFF_U + id] = u * psi;
}

// ---------------- write content weights (cosine + softmax) ----------------
__global__ void k_wc(const float* __restrict__ memory, float* __restrict__ ws){
  __shared__ float kk[128];
  __shared__ float sbuf[1024];
  __shared__ float red[256];
  int b = blockIdx.x, tid = threadIdx.x;
  if (tid < 128) kk[tid] = ws[OFF_WKEY + b*128 + tid];
  __syncthreads();
  float kn2 = 0.0f;
  for (int w = 0; w < 128; ++w) kn2 += kk[w]*kk[w];
  float kn = sqrtf(kn2);
  float beta = ws[OFF_SMALL + b];
  for (int n = tid; n < Nn; n += 256) {
    const float* m = memory + ((size_t)b*Nn + n)*Wd;
    float dot = 0.0f, mn2 = 0.0f;
    for (int w = 0; w < 128; ++w) { float mv = m[w]; dot += kk[w]*mv; mn2 += mv*mv; }
    sbuf[n] = beta * (dot / (kn*sqrtf(mn2) + EPSv));
  }
  __syncthreads();
  float lm = -INFINITY;
  for (int n = tid; n < Nn; n += 256) lm = fmaxf(lm, sbuf[n]);
  red[tid] = lm; __syncthreads();
  for (int o = 128; o > 0; o >>= 1) { if (tid < o) red[tid] = fmaxf(red[tid], red[tid+o]); __syncthreads(); }
  float mx = red[0]; __syncthreads();
  float ls = 0.0f;
  for (int n = tid; n < Nn; n += 256) { float e = expf(sbuf[n]-mx); sbuf[n] = e; ls += e; }
  red[tid] = ls; __syncthreads();
  for (int o = 128; o > 0; o >>= 1) { if (tid < o) red[tid] += red[tid+o]; __syncthreads(); }
  float inv = 1.0f / red[0];
  for (int n = tid; n < Nn; n += 256) ws[OFF_WC + (size_t)b*Nn + n] = sbuf[n]*inv;
}

// ---------------- allocation (bitonic sort) + wl_write ----------------
__global__ __launch_bounds__(1024) void k_alloc(float* __restrict__ ws){
  __shared__ float key[1024];
  __shared__ int   idx[1024];
  __shared__ float excl[1024];
  int b = blockIdx.x, tid = threadIdx.x;
  key[tid] = ws[OFF_U + (size_t)b*Nn + tid];
  idx[tid] = tid;
  __syncthreads();
  for (int k = 2; k <= 1024; k <<= 1)
    for (int j = k >> 1; j > 0; j >>= 1) {
      int i = tid, ixj = i ^ j;
      if (ixj > i) {
        bool up = ((i & k) == 0);
        float ki = key[i], kj = key[ixj];
        bool sw = up ? (ki > kj) : (ki < kj);
        if (sw) { key[i] = kj; key[ixj] = ki; int t = idx[i]; idx[i] = idx[ixj]; idx[ixj] = t; }
      }
      __syncthreads();
    }
  if (tid == 0) {
    float run = 1.0f;
    for (int n = 0; n < 1024; ++n) { excl[n] = run; run *= key[n]; }
  }
  __syncthreads();
  float wg = ws[OFF_SMALL + 64 + b], ag = ws[OFF_SMALL + 32 + b];
  float a_s = (1.0f - key[tid]) * excl[tid];
  int n = idx[tid];
  float wcv = ws[OFF_WC + (size_t)b*Nn + n];
  ws[OFF_WLW + (size_t)b*Nn + n] = wg*(ag*a_s + (1.0f-ag)*wcv);
}

// ---------------- memory write (erase+add) ----------------
__global__ void k_memnew(const float* __restrict__ memory, float* __restrict__ ws){
  int id = blockIdx.x*256 + threadIdx.x;   // B*N*W/4
  int w4 = (id & 31)*4;
  int n  = (id >> 5) & 1023;
  int b  = id >> 15;
  float wlw = ws[OFF_WLW + b*Nn + n];
  const float4 m = *(const float4*)(memory + (size_t)id*4);
  const float4 er = *(const float4*)(ws + OFF_ERASE + b*128 + w4);
  const float4 ad = *(const float4*)(ws + OFF_ADD   + b*128 + w4);
  float4 o;
  o.x = m.x*(1.0f - wlw*er.x) + wlw*ad.x;
  o.y = m.y*(1.0f - wlw*er.y) + wlw*ad.y;
  o.z = m.z*(1.0f - wlw*er.z) + wlw*ad.z;
  o.w = m.w*(1.0f - wlw*er.w) + wlw*ad.w;
  *(float4*)(ws + OFF_MEMNEW + (size_t)id*4) = o;
}

// ---------------- read content weights (4 heads, one memory pass) ----------------
__global__ void k_rc(float* __restrict__ ws){
  __shared__ float rk[4*128];
  __shared__ float sbuf[4*1024];
  __shared__ float red[256];
  int b = blockIdx.x, tid = threadIdx.x;
  for (int t = tid; t < 512; t += 256) rk[t] = ws[OFF_RKEY + b*512 + t];
  __syncthreads();
  float kn[4], beta[4];
  for (int r = 0; r < 4; ++r) {
    float s = 0.0f;
    for (int w = 0; w < 128; ++w) { float v = rk[r*128+w]; s += v*v; }
    kn[r] = sqrtf(s);
    beta[r] = ws[OFF_SMALL + 128 + b*4 + r];
  }
  for (int n = tid; n < Nn; n += 256) {
    const float* m = ws + OFF_MEMNEW + ((size_t)b*Nn + n)*Wd;
    float d0=0,d1=0,d2=0,d3=0,mn2=0;
    for (int w = 0; w < 128; ++w) {
      float mv = m[w]; mn2 += mv*mv;
      d0 += rk[w]*mv; d1 += rk[128+w]*mv; d2 += rk[256+w]*mv; d3 += rk[384+w]*mv;
    }
    float mn = sqrtf(mn2);
    sbuf[n]      = beta[0]*d0/(kn[0]*mn + EPSv);
    sbuf[1024+n] = beta[1]*d1/(kn[1]*mn + EPSv);
    sbuf[2048+n] = beta[2]*d2/(kn[2]*mn + EPSv);
    sbuf[3072+n] = beta[3]*d3/(kn[3]*mn + EPSv);
  }
  __syncthreads();
  for (int r = 0; r < 4; ++r) {
    float lm = -INFINITY;
    for (int n = tid; n < Nn; n += 256) lm = fmaxf(lm, sbuf[r*1024+n]);
    red[tid] = lm; __syncthreads();
    for (int o = 128; o > 0; o >>= 1) { if (tid < o) red[tid] = fmaxf(red[tid], red[tid+o]); __syncthreads(); }
    float mx = red[0]; __syncthreads();
    float ls = 0.0f;
    for (int n = tid; n < Nn; n += 256) { float e = expf(sbuf[r*1024+n]-mx); sbuf[r*1024+n] = e; ls += e; }
    red[tid] = ls; __syncthreads();
    for (int o = 128; o > 0; o >>= 1) { if (tid < o) red[tid] += red[tid+o]; __syncthreads(); }
    float inv = 1.0f / red[0]; __syncthreads();
    for (int n = tid; n < Nn; n += 256)
      ws[OFF_RC + ((size_t)b*4+r)*Nn + n] = sbuf[r*1024+n]*inv;
  }
}

// ---------------- fused link_new + fwd/bwd matvecs (single pass over link) ----------------
__global__ __launch_bounds__(256) void k_link(const float* __restrict__ link,
    const float* __restrict__ preced, const float* __restrict__ wlpr,
    float* __restrict__ ws){
  __shared__ float wRow[64];
  __shared__ float qRow[4][64];
  __shared__ float waveF[8][4][64];
  int b  = blockIdx.x >> 4;
  int it = blockIdx.x & 15;
  int i0 = it*64;
  int tid = threadIdx.x;
  const float* wlw = ws + OFF_WLW + (size_t)b*Nn;
  if (tid < 64) wRow[tid] = wlw[i0 + tid];
  { int r = tid >> 6, ii = tid & 63; qRow[r][ii] = wlpr[((size_t)b*4+r)*Nn + i0 + ii]; }
  __syncthreads();
  int j0 = tid*4;
  float wj[4], pj[4], qj[4][4];
  for (int jj = 0; jj < 4; ++jj) { wj[jj] = wlw[j0+jj]; pj[jj] = preced[(size_t)b*Nn + j0+jj]; }
  for (int r = 0; r < 4; ++r)
    for (int jj = 0; jj < 4; ++jj) qj[r][jj] = wlpr[((size_t)b*4+r)*Nn + j0+jj];
  float bwdReg[4][4] = {};
  int wave = tid >> 5, lane = tid & 31;
  const float* Lrow = link + (size_t)b*Nn*Nn + (size_t)i0*Nn + j0;
  for (int i = 0; i < 64; ++i) {
    float wi = wRow[i];
    float qi0 = qRow[0][i], qi1 = qRow[1][i], qi2 = qRow[2][i], qi3 = qRow[3][i];
    float4 L = *(const float4*)(Lrow + (size_t)i*Nn);
    float La[4] = {L.x, L.y, L.z, L.w};
    float f0=0,f1=0,f2=0,f3=0;
    for (int jj = 0; jj < 4; ++jj) {
      float ln = (1.0f - wi - wj[jj])*La[jj] + wi*pj[jj];
      if (i0+i == j0+jj) ln = 0.0f;                 // zero diagonal
      f0 += ln*qj[0][jj]; f1 += ln*qj[1][jj]; f2 += ln*qj[2][jj]; f3 += ln*qj[3][jj];
      bwdReg[0][jj] += ln*qi0; bwdReg[1][jj] += ln*qi1;
      bwdReg[2][jj] += ln*qi2; bwdReg[3][jj] += ln*qi3;
    }
    for (int off = 16; off > 0; off >>= 1) {
      f0 += __shfl_down(f0, off); f1 += __shfl_down(f1, off);
      f2 += __shfl_down(f2, off); f3 += __shfl_down(f3, off);
    }
    if (lane == 0) { waveF[wave][0][i]=f0; waveF[wave][1][i]=f1; waveF[wave][2][i]=f2; waveF[wave][3][i]=f3; }
  }
  __syncthreads();
  { // fwd: block owns full rows -> direct store, no atomics
    int r = tid >> 6, i = tid & 63;
    float s = 0.0f;
    for (int wv = 0; wv < 8; ++wv) s += waveF[wv][r][i];
    ws[OFF_FWD + ((size_t)b*4+r)*Nn + i0 + i] = s;
  }
  for (int r = 0; r < 4; ++r)
    for (int jj = 0; jj < 4; ++jj)
      atomicAdd(ws + OFF_BWD + ((size_t)b*4+r)*Nn + j0+jj, bwdReg[r][jj]);
}

// ---------------- pi softmax combine -> wl_read ----------------
__global__ void k_wlread(float* __restrict__ ws){
  int id = blockIdx.x*256 + threadIdx.x;  // B*4*N
  int r = (id >> 10) & 3;
  int b = id >> 12;
  const float* pr = ws + OFF_SMALL + 384 + b*12 + r*3;
  float p0 = pr[0], p1 = pr[1], p2 = pr[2];
  float mx = fmaxf(p0, fmaxf(p1, p2));
  float e0 = expf(p0-mx), e1 = expf(p1-mx), e2 = expf(p2-mx);
  float inv = 1.0f/(e0+e1+e2);
  float bw = ws[OFF_BWD + id], rc = ws[OFF_RC + id], fw = ws[OFF_FWD + id];
  ws[OFF_WLREAD + id] = (e0*bw + e1*rc + e2*fw)*inv;
}

// ---------------- readout GEMM via WMMA: read_vec = wl_read x mem_new ----------------
__global__ __launch_bounds__(32) void k_readvec(const float* __restrict__ ws, float* __restrict__ out){
  int b  = blockIdx.x >> 3;
  int ct = blockIdx.x & 7;
  int c0 = ct*16;
  int lane = threadIdx.x, lo = lane & 15, hi = lane >> 4;
  const float* wlr = ws + OFF_WLREAD + (size_t)b*4*Nn;
  const float* M   = ws + OFF_MEMNEW + (size_t)b*Nn*Wd + c0 + lo;
  v8f acc = {0,0,0,0,0,0,0,0};
  for (int k = 0; k < Nn; k += 4) {
    int ka = k + 2*hi;
    v2f a, bb;
    a.x = (lo < 4) ? wlr[(size_t)lo*Nn + ka]     : 0.0f;   // rows 4..15 padded with zeros
    a.y = (lo < 4) ? wlr[(size_t)lo*Nn + ka + 1] : 0.0f;
    bb.x = M[(size_t)ka*Wd];
    bb.y = M[(size_t)(ka+1)*Wd];
    acc = __builtin_amdgcn_wmma_f32_16x16x4_f32(false, a, false, bb, (short)0, acc, false, false);
  }
  if (hi == 0) {
    for (int v = 0; v < 4; ++v)
      out[((size_t)b*4 + v)*Wd + c0 + lo] = acc[v];
  }
}

extern "C" void kernel_launch(void* const* d_in, const int* in_sizes, int n_in,
                              void* d_out, int out_size, void* d_ws, size_t ws_size,
                              hipStream_t stream) {
  (void)in_sizes; (void)n_in; (void)out_size; (void)ws_size;
  const float* hidden = (const float*)d_in[0];
  const float* memory = (const float*)d_in[1];
  const float* usage  = (const float*)d_in[2];
  const float* link   = (const float*)d_in[3];
  const float* preced = (const float*)d_in[4];
  const float* wlpw   = (const float*)d_in[5];
  const float* wlpr   = (const float*)d_in[6];
  const float* Wkey   = (const float*)d_in[7];
  const float* Wbeta  = (const float*)d_in[8];
  const float* Walloc = (const float*)d_in[9];
  const float* Wwrite = (const float*)d_in[10];
  const float* Werase = (const float*)d_in[11];
  const float* Wadd   = (const float*)d_in[12];
  const float* Wrkey  = (const float*)d_in[13];
  const float* Wrbeta = (const float*)d_in[14];
  const float* Wrfree = (const float*)d_in[15];
  const float* Wrpi   = (const float*)d_in[16];
  float* ws  = (float*)d_ws;
  float* out = (float*)d_out;

  k_zero      <<<512, 256, 0, stream>>>(ws);
  k_proj_wmma <<<112,  32, 0, stream>>>(hidden, Wkey, Werase, Wadd, Wrkey, ws);
  k_proj_small<<<Bsz*23, 256, 0, stream>>>(hidden, Wbeta, Walloc, Wwrite, Wrbeta, Wrfree, Wrpi, ws);
  k_usage     <<<128, 256, 0, stream>>>(usage, wlpw, wlpr, ws);
  k_wc        <<<Bsz, 256, 0, stream>>>(memory, ws);
  k_alloc     <<<Bsz, 1024, 0, stream>>>(ws);
  k_memnew    <<<4096, 256, 0, stream>>>(memory, ws);
  k_rc        <<<Bsz, 256, 0, stream>>>(ws);
  k_link      <<<Bsz*16, 256, 0, stream>>>(link, preced, wlpr, ws);
  k_wlread    <<<512, 256, 0, stream>>>(ws);
  k_readvec   <<<Bsz*8, 32, 0, stream>>>(ws, out);
}